// Model_80642305950257
// MI455X (gfx1250) — compile-verified
//
#include <hip/hip_runtime.h>
#include <math.h>

#ifndef M_PI
#define M_PI 3.14159265358979323846
#endif

typedef __attribute__((ext_vector_type(2))) float v2f;
typedef __attribute__((ext_vector_type(8))) float v8f;

static constexpr long E3  = 262144;   // 64^3 spatial elements per channel
static constexpr long SZ2 = 63 * 63;  // 3969

// ---------------------------------------------------------------------------
// Generic strided batched GEMM on f32 WMMA (V_WMMA_F32_16X16X4_F32).
//   D[b1,b2][m,n] (=|+=) alpha * sum_k A[b1,b2][m,k] * B[b1,b2][k,n]
// One wave = one 32x32 output tile: 2x2 register tiling with four 16x16
// accumulators sharing two A fragments and two B fragments per K-step
// (8 loads -> 4 WMMAs). Out-of-range rows/cols are CLAMPED (their products
// only reach accumulator lanes discarded at store time), so the hot loop has
// unconditional loads, pointer-increment addressing, and no EXEC churn.
// Only the K tail needs explicit zero-fill.
// A fragment (16x4): lane L: row=L%16; vgpr0=K(k+2*(L/16)), vgpr1=K(+1)
// B fragment (4x16): lane L: col=L%16; vgpr0=K(k+2*(L/16)), vgpr1=K(+1)
// C/D (16x16): vgpr v, lane L: row=v+8*(L/16), col=L%16
// ---------------------------------------------------------------------------
__global__ void k_gemm(const float* __restrict__ A, long aR, long aC, long aB1, long aB2,
                       const float* __restrict__ B, long bR, long bC, long bB1, long bB2,
                       float* __restrict__ D, long dR, long dC, long dB1, long dB2,
                       int M, int N, int K, float alpha, int accum)
{
    const int lane    = threadIdx.x & 31;
    const int wave    = threadIdx.x >> 5;
    const int tilesM2 = (M + 31) >> 5;
    const int tilesN2 = (N + 31) >> 5;
    const int tileId  = blockIdx.x * (blockDim.x >> 5) + wave;
    if (tileId >= tilesM2 * tilesN2) return;  // wave-uniform: EXEC all-ones for WMMA
    const int tm = (tileId / tilesN2) * 2;
    const int tn = (tileId % tilesN2) * 2;

    const float* Ap = A + (long)blockIdx.y * aB1 + (long)blockIdx.z * aB2;
    const float* Bp = B + (long)blockIdx.y * bB1 + (long)blockIdx.z * bB2;
    float*       Dp = D + (long)blockIdx.y * dB1 + (long)blockIdx.z * dB2;

    const int hl  = lane >> 4;     // 0 or 1
    const int l16 = lane & 15;

    // clamped fragment coordinates (see note above)
    int arow0 = tm * 16 + l16;      if (arow0 > M - 1) arow0 = M - 1;
    int arow1 = tm * 16 + 16 + l16; if (arow1 > M - 1) arow1 = M - 1;
    int bcol0 = tn * 16 + l16;      if (bcol0 > N - 1) bcol0 = N - 1;
    int bcol1 = tn * 16 + 16 + l16; if (bcol1 > N - 1) bcol1 = N - 1;

    const float* ap0 = Ap + (long)arow0 * aR + (long)(2 * hl) * aC;
    const float* ap1 = Ap + (long)arow1 * aR + (long)(2 * hl) * aC;
    const float* bp0 = Bp + (long)bcol0 * bC + (long)(2 * hl) * bR;
    const float* bp1 = Bp + (long)bcol1 * bC + (long)(2 * hl) * bR;
    const long aStep = 4 * aC;
    const long bStep = 4 * bR;

    v8f acc00 = {0.f, 0.f, 0.f, 0.f, 0.f, 0.f, 0.f, 0.f};
    v8f acc01 = {0.f, 0.f, 0.f, 0.f, 0.f, 0.f, 0.f, 0.f};
    v8f acc10 = {0.f, 0.f, 0.f, 0.f, 0.f, 0.f, 0.f, 0.f};
    v8f acc11 = {0.f, 0.f, 0.f, 0.f, 0.f, 0.f, 0.f, 0.f};

    const int K4 = K & ~3;
    for (int k = 0; k < K4; k += 4) {
        v2f a0, a1, b0, b1;
        a0.x = ap0[0];  a0.y = ap0[aC];
        a1.x = ap1[0];  a1.y = ap1[aC];
        b0.x = bp0[0];  b0.y = bp0[bR];
        b1.x = bp1[0];  b1.y = bp1[bR];
        acc00 = __builtin_amdgcn_wmma_f32_16x16x4_f32(false, a0, false, b0, (short)0, acc00,
                                                      false, false);
        acc01 = __builtin_amdgcn_wmma_f32_16x16x4_f32(false, a0, false, b1, (short)0, acc01,
                                                      false, false);
        acc10 = __builtin_amdgcn_wmma_f32_16x16x4_f32(false, a1, false, b0, (short)0, acc10,
                                                      false, false);
        acc11 = __builtin_amdgcn_wmma_f32_16x16x4_f32(false, a1, false, b1, (short)0, acc11,
                                                      false, false);
        ap0 += aStep; ap1 += aStep; bp0 += bStep; bp1 += bStep;
    }
    if (K4 < K) {   // K tail: zero-fill beyond K
        const int ka = K4 + 2 * hl;
        v2f a0, a1, b0, b1;
        a0.x = 0.f; a0.y = 0.f; a1.x = 0.f; a1.y = 0.f;
        b0.x = 0.f; b0.y = 0.f; b1.x = 0.f; b1.y = 0.f;
        if (ka < K)     { a0.x = ap0[0];  a1.x = ap1[0];  b0.x = bp0[0];  b1.x = bp1[0];  }
        if (ka + 1 < K) { a0.y = ap0[aC]; a1.y = ap1[aC]; b0.y = bp0[bR]; b1.y = bp1[bR]; }
        acc00 = __builtin_amdgcn_wmma_f32_16x16x4_f32(false, a0, false, b0, (short)0, acc00,
                                                      false, false);
        acc01 = __builtin_amdgcn_wmma_f32_16x16x4_f32(false, a0, false, b1, (short)0, acc01,
                                                      false, false);
        acc10 = __builtin_amdgcn_wmma_f32_16x16x4_f32(false, a1, false, b0, (short)0, acc10,
                                                      false, false);
        acc11 = __builtin_amdgcn_wmma_f32_16x16x4_f32(false, a1, false, b1, (short)0, acc11,
                                                      false, false);
    }

#pragma unroll
    for (int v = 0; v < 8; ++v) {
        const int r0 = tm * 16 + v + 8 * hl;
        const int r1 = r0 + 16;
        const int c0 = tn * 16 + l16;
        const int c1 = c0 + 16;
        if (r0 < M) {
            if (c0 < N) {
                const long idx = (long)r0 * dR + (long)c0 * dC;
                const float val = alpha * acc00[v];
                if (accum) Dp[idx] += val; else Dp[idx] = val;
            }
            if (c1 < N) {
                const long idx = (long)r0 * dR + (long)c1 * dC;
                const float val = alpha * acc01[v];
                if (accum) Dp[idx] += val; else Dp[idx] = val;
            }
        }
        if (r1 < M) {
            if (c0 < N) {
                const long idx = (long)r1 * dR + (long)c0 * dC;
                const float val = alpha * acc10[v];
                if (accum) Dp[idx] += val; else Dp[idx] = val;
            }
            if (c1 < N) {
                const long idx = (long)r1 * dR + (long)c1 * dC;
                const float val = alpha * acc11[v];
                if (accum) Dp[idx] += val; else Dp[idx] = val;
            }
        }
    }
}

// ---------------------------------------------------------------------------
// Constant tables: log-factorials, quadrature weights, centered DFT matrices.
//   WC[a][j] = exp(-2*pi*i * a * (j-31) / 64)          (fft + fftshift-crop; == conj(PHASE))
//   VC[j][a] = exp(+2*pi*i * a * (j-31) / 64) / 64     (pad + ifftshift + ifft)
// ---------------------------------------------------------------------------
__global__ void k_tables(double* __restrict__ lf, float* __restrict__ qw,
                         float* __restrict__ WCre, float* __restrict__ WCim,
                         float* __restrict__ VCre, float* __restrict__ VCim)
{
    const int t = threadIdx.x;
    if (t < 65) {
        double s = 0.0;
        for (int j = 1; j <= t; ++j) s += log((double)j);
        lf[t] = s;
    }
    if (t < 64) {
        double s = 0.0;
        for (int k = 0; k < 32; ++k)
            s += 1.0 / (2 * k + 1) * sin((2 * t + 1) * (2 * k + 1) * M_PI / 128.0);
        qw[t] = (float)(2.0 / 32.0 * sin(M_PI * (2 * t + 1) / 128.0) * s);
    }
    for (int idx = t; idx < 64 * 63; idx += blockDim.x) {
        int a = idx / 63, j = idx % 63;
        double ang = -2.0 * M_PI * (double)a * (double)(j - 31) / 64.0;
        WCre[idx] = (float)cos(ang);
        WCim[idx] = (float)sin(ang);
    }
    for (int idx = t; idx < 63 * 64; idx += blockDim.x) {
        int j = idx / 64, a = idx % 64;
        double ang = 2.0 * M_PI * (double)a * (double)(j - 31) / 64.0;
        VCre[idx] = (float)(cos(ang) / 64.0);
        VCim[idx] = (float)(sin(ang) / 64.0);
    }
}

// ---------------------------------------------------------------------------
// Wigner little-d tables for the 64 quadrature betas (+ equator).
//   WDF[z][l][mp][m] = qw[z] * d^l_{mp,m}(beta_z)         (forward, WD_FWD)
//   WDI[z][l][mp][m] = (2l+1)/2 * d^l_{mp,m}(beta_z)      (inverse, WD_INV)
//   DEQ[l][mp][m]    = d^l_{mp,m}(pi/2)
// ---------------------------------------------------------------------------
__global__ void k_wigner(const double* __restrict__ lf, const float* __restrict__ qw,
                         float* __restrict__ WDF, float* __restrict__ WDI,
                         float* __restrict__ DEQ)
{
    int t = blockIdx.x * blockDim.x + threadIdx.x;
    if (t >= 3969) return;
    const int mpI = t / 63, mI = t % 63;
    const int mp = mpI - 31, m = mI - 31;
    const int l  = blockIdx.y;
    const int bi = blockIdx.z;
    const double beta = (bi < 64) ? ((bi + 0.5) / 64.0) * M_PI : (M_PI * 0.5);

    double val = 0.0;
    const int amp = mp < 0 ? -mp : mp;
    const int am  = m  < 0 ? -m  : m;
    if (amp <= l && am <= l) {
        const double ch = cos(beta * 0.5), sh = sin(beta * 0.5);
        const double lc = log(ch), ls = log(sh);
        const int k0 = (m - mp) > 0 ? (m - mp) : 0;
        const int k1 = (l + m) < (l - mp) ? (l + m) : (l - mp);
        const double pref = 0.5 * (lf[l + mp] + lf[l - mp] + lf[l + m] + lf[l - m]);
        for (int k = k0; k <= k1; ++k) {
            const double logc = pref - (lf[l + m - k] + lf[k] + lf[mp - m + k] + lf[l - mp - k]);
            const double sgn = ((mp - m + k) & 1) ? -1.0 : 1.0;
            const double ca = (double)(2 * l + m - mp - 2 * k);
            const double sb = (double)(mp - m + 2 * k);
            val += sgn * exp(logc + ca * lc + sb * ls);
        }
    }
    if (bi < 64) {
        const long idx = (((long)bi * 32 + l) * 63 + mpI) * 63 + mI;
        WDF[idx] = (float)(qw[bi] * val);
        WDI[idx] = (float)((2.0 * l + 1.0) * 0.5 * val);
    } else {
        const long idx = ((long)l * 63 + mpI) * 63 + mI;
        DEQ[idx] = (float)val;
    }
}

// z[b,o,l,m,n] = sum_i F[b,i,l,m] * Kh[i,o,n] * D0EQ[l,n]   (i = 3, complex)
__global__ void k_s2combine(const float* __restrict__ Fz, const float* __restrict__ Kh,
                            const float* __restrict__ DEQ, float* __restrict__ zz)
{
    const long PF = 12096, PK = 4725, PZ = 6350400;
    long t = (long)blockIdx.x * blockDim.x + threadIdx.x;
    if (t >= PZ) return;
    int n = (int)(t % 63); long r = t / 63;
    int m = (int)(r % 63); r /= 63;
    int l = (int)(r % 32); r /= 32;
    int o = (int)(r % 25);
    int b = (int)(r / 25);
    const float d0 = DEQ[((long)l * 63 + n) * 63 + 31];   // D0EQ[l,n]
    float zr = 0.f, zi = 0.f;
    for (int i = 0; i < 3; ++i) {
        const long fi = (((long)b * 3 + i) * 32 + l) * 63 + m;
        const float fr = Fz[fi], fim = Fz[fi + PF];
        const long ki = ((long)i * 25 + o) * 63 + n;
        const float kr = Kh[ki], kim = Kh[ki + PK];
        zr += fr * kr - fim * kim;
        zi += fr * kim + fim * kr;
    }
    zz[t]      = zr * d0;
    zz[t + PZ] = zi * d0;
}

__global__ void k_biasadd(float* __restrict__ x, const float* __restrict__ bias,
                          int C, long total)
{
    long t = (long)blockIdx.x * blockDim.x + threadIdx.x;
    if (t >= total) return;
    x[t] += bias[(int)((t / E3) % C)];
}

__global__ void k_bn_stats(const float* __restrict__ x, long chStride, long bStride,
                           int Bn, long n, float* __restrict__ mean, float* __restrict__ rstd)
{
    __shared__ double s1[256], s2[256];
    double a = 0.0, q = 0.0;
    for (int b = 0; b < Bn; ++b) {
        const float* p = x + (long)b * bStride + (long)blockIdx.x * chStride;
        for (long i = threadIdx.x; i < n; i += blockDim.x) {
            const double v = p[i];
            a += v; q += v * v;
        }
    }
    s1[threadIdx.x] = a; s2[threadIdx.x] = q;
    __syncthreads();
    for (int o = 128; o > 0; o >>= 1) {
        if ((int)threadIdx.x < o) { s1[threadIdx.x] += s1[threadIdx.x + o]; s2[threadIdx.x] += s2[threadIdx.x + o]; }
        __syncthreads();
    }
    if (threadIdx.x == 0) {
        const double cnt = (double)Bn * (double)n;
        const double mu  = s1[0] / cnt;
        const double var = s2[0] / cnt - mu * mu;
        mean[blockIdx.x] = (float)mu;
        rstd[blockIdx.x] = (float)(1.0 / sqrt(var + 1e-5));
    }
}

// dst[b, cofs+c, :] = relu((x[b,c,:]-mean[c])*rstd[c]*g[c]+bb[c])
__global__ void k_bn_apply(const float* __restrict__ x, float* __restrict__ dst,
                           const float* __restrict__ mean, const float* __restrict__ rstd,
                           const float* __restrict__ g, const float* __restrict__ bb,
                           int C, int Cdst, int cofs, long total)
{
    long t = (long)blockIdx.x * blockDim.x + threadIdx.x;
    if (t >= total) return;
    const long e = t % E3;
    long r = t / E3;
    const int c = (int)(r % C);
    const int b = (int)(r / C);
    const float v = (x[t] - mean[c]) * rstd[c] * g[c] + bb[c];
    dst[((long)(b * Cdst + cofs + c)) * E3 + e] = fmaxf(v, 0.f);
}

__global__ void k_meanlast(const float* __restrict__ h, float* __restrict__ out, long total)
{
    long t = (long)blockIdx.x * blockDim.x + threadIdx.x;
    if (t >= total) return;
    const float* p = h + t * 64;
    float s = 0.f;
    for (int a = 0; a < 64; ++a) s += p[a];
    out[t] = s * (1.f / 64.f);
}

// ------------------------------- host side ---------------------------------
namespace {

struct Ctx {
    hipStream_t st;
    float *WCre, *WCim, *VCre, *VCim, *WDF, *WDI, *DEQ;
    float *RA, *RB, *RC;           // arena regions
    float *mean, *rstd;
};

inline void gemm(hipStream_t st,
                 const float* A, long aR, long aC, long aB1, long aB2,
                 const float* B, long bR, long bC, long bB1, long bB2,
                 float* D, long dR, long dC, long dB1, long dB2,
                 int M, int N, int K, int nb1, int nb2, float alpha, int accum)
{
    const int tiles  = ((M + 31) / 32) * ((N + 31) / 32);
    const int blocks = (tiles + 7) / 8;
    dim3 g((unsigned)blocks, (unsigned)nb1, (unsigned)nb2);
    k_gemm<<<g, 256, 0, st>>>(A, aR, aC, aB1, aB2, B, bR, bC, bB1, bB2,
                              D, dR, dC, dB1, dB2, M, N, K, alpha, accum);
}

// complex = complex * complex  (4 real GEMMs)
inline void cgemm_cc(hipStream_t st,
                     const float* Are, const float* Aim, long aR, long aC, long aB1, long aB2,
                     const float* Bre, const float* Bim, long bR, long bC, long bB1, long bB2,
                     float* Dre, float* Dim, long dR, long dC, long dB1, long dB2,
                     int M, int N, int K, int nb1, int nb2, float alpha)
{
    gemm(st, Are, aR, aC, aB1, aB2, Bre, bR, bC, bB1, bB2, Dre, dR, dC, dB1, dB2, M, N, K, nb1, nb2,  alpha, 0);
    gemm(st, Aim, aR, aC, aB1, aB2, Bim, bR, bC, bB1, bB2, Dre, dR, dC, dB1, dB2, M, N, K, nb1, nb2, -alpha, 1);
    gemm(st, Are, aR, aC, aB1, aB2, Bim, bR, bC, bB1, bB2, Dim, dR, dC, dB1, dB2, M, N, K, nb1, nb2,  alpha, 0);
    gemm(st, Aim, aR, aC, aB1, aB2, Bre, bR, bC, bB1, bB2, Dim, dR, dC, dB1, dB2, M, N, K, nb1, nb2,  alpha, 1);
}

// so3_synth: z (planar complex [b][Co][32][63][63]) -> real out [b][Co][64][64][64]
// Uses RA for Xt, RB for Y (z may alias RB: z is dead before Y is written).
void so3_synth(const Ctx& c, const float* zz, int Co, float* outReal)
{
    const long PZ  = (long)2 * Co * 32 * SZ2;
    const long PXT = (long)2 * Co * 64 * SZ2;
    const long PY  = (long)2 * Co * 64 * 63 * 64;
    float* Xt = c.RA;
    float* Y  = c.RB;
    // E: Xt[b,c,z,m,n] = sum_l WDI[z,l,m,n] * z[b,c,l,m,n]   (batch over (m,n))
    for (int p = 0; p < 2; ++p)
        gemm(c.st, zz + p * PZ, (long)32 * SZ2, SZ2, 1, 0,
                   c.WDI, SZ2, (long)32 * SZ2, 1, 0,
                   Xt + p * PXT, (long)64 * SZ2, SZ2, 1, 0,
                   2 * Co, 64, 32, 3969, 1, 1.f, 0);
    // F: Y[...,m,a] = sum_n Xt[...,m,n] * VC[n,a]   (complex x complex)
    cgemm_cc(c.st, Xt, Xt + PXT, 63, 1, 0, 0,
                   c.VCre, c.VCim, 64, 1, 0, 0,
                   Y, Y + PY, 64, 1, 0, 0,
                   2 * Co * 64 * 63, 64, 63, 1, 1, 1.f);
    // G: out[a_m,a_n] = Re( sum_m VC[m,a_m] * Y[m,a_n] )   (batch over (b,c,z))
    gemm(c.st, c.VCre, 1, 64, 0, 0, Y,      64, 1, (long)63 * 64, 0,
         outReal, 64, 1, 4096, 0, 64, 64, 63, 2 * Co * 64, 1,  1.f, 0);
    gemm(c.st, c.VCim, 1, 64, 0, 0, Y + PY, 64, 1, (long)63 * 64, 0,
         outReal, 64, 1, 4096, 0, 64, 64, 63, 2 * Co * 64, 1, -1.f, 1);
}

// so3_conv: Hin real [b][Ci][64^3], Kw [Ci][Co][64] -> outReal [b][Co][64^3]
void so3_conv(const Ctx& c, const float* Hin, int Ci, const float* Kw, int Co,
              float scale, float* Kh, float* outReal, const float* bias)
{
    const long PT  = (long)2 * Ci * 64 * 64 * 63;
    const long PX  = (long)2 * Ci * 64 * SZ2;
    const long PF  = (long)2 * Ci * 32 * SZ2;
    const long PKH = (long)Ci * Co * 63;
    const long PZ  = (long)2 * Co * 32 * SZ2;
    float* T1 = c.RA;
    float* Xc = c.RB;
    float* Ff = c.RC;
    float* CC = c.RA;   // reuses T1 (dead)
    float* zz = c.RB;   // reuses Xc (dead)

    // Kh[i,o,n] = scale * sum_p Kw[i,o,p] * WC[p,n]
    gemm(c.st, Kw, 64, 1, 0, 0, c.WCre, 63, 1, 0, 0, Kh,       63, 1, 0, 0, Ci * Co, 63, 64, 1, 1, scale, 0);
    gemm(c.st, Kw, 64, 1, 0, 0, c.WCim, 63, 1, 0, 0, Kh + PKH, 63, 1, 0, 0, Ci * Co, 63, 64, 1, 1, scale, 0);
    // H: T1[b,c,z,am,n] = sum_an x[...,an] * WC[an,n]   (real x complex)
    const int rowsH = 2 * Ci * 64 * 64;
    gemm(c.st, Hin, 64, 1, 0, 0, c.WCre, 63, 1, 0, 0, T1,      63, 1, 0, 0, rowsH, 63, 64, 1, 1, 1.f, 0);
    gemm(c.st, Hin, 64, 1, 0, 0, c.WCim, 63, 1, 0, 0, T1 + PT, 63, 1, 0, 0, rowsH, 63, 64, 1, 1, 1.f, 0);
    // I: Xc[m,n] = sum_am WC[am,m] * T1[am,n]   (batch (b,c,z), complex x complex)
    cgemm_cc(c.st, c.WCre, c.WCim, 1, 63, 0, 0,
                   T1, T1 + PT, 63, 1, (long)64 * 63, 0,
                   Xc, Xc + PX, 63, 1, SZ2, 0,
                   63, 63, 64, 2 * Ci * 64, 1, 1.f);
    // J: F[b,c,l,m,n] = sum_z WDF[z,l,m,n] * Xc[b,c,z,m,n]   (batch (m,n))
    for (int p = 0; p < 2; ++p)
        gemm(c.st, Xc + p * PX, (long)64 * SZ2, SZ2, 1, 0,
                   c.WDF, (long)32 * SZ2, SZ2, 1, 0,
                   Ff + p * PF, (long)32 * SZ2, SZ2, 1, 0,
                   2 * Ci, 32, 64, 3969, 1, 1.f, 0);
    // K: C[b,i,l,m,n] = sum_k F[b,i,l,m,k] * DEQ[l,n,k]   (batch1=l, batch2=(b,i))
    for (int p = 0; p < 2; ++p)
        gemm(c.st, Ff + p * PF, 63, 1, SZ2, (long)32 * SZ2,
                   c.DEQ, 1, 63, SZ2, 0,
                   CC + p * PF, 63, 1, SZ2, (long)32 * SZ2,
                   63, 63, 63, 32, 2 * Ci, 1.f, 0);
    // L: z[b,o,l,m,n] = sum_i Kh[i,o,n] * C[b,i,l,m,n]   (batch1=n, batch2=b)
    cgemm_cc(c.st, CC, CC + PF, 63, (long)32 * SZ2, 1, (long)Ci * 32 * SZ2,
                   Kh, Kh + PKH, (long)Co * 63, 63, 1, 0,
                   zz, zz + PZ, 63, (long)32 * SZ2, 1, (long)Co * 32 * SZ2,
                   2016, Co, Ci, 63, 2, 1.f);
    so3_synth(c, zz, Co, outReal);
    const long tot = (long)2 * Co * E3;
    k_biasadd<<<(unsigned)((tot + 255) / 256), 256, 0, c.st>>>(outReal, bias, Co, tot);
}

void bn_block(const Ctx& c, const float* src, int C, int cofs, float* dst, int Cdst,
              const float* g, const float* bb)
{
    k_bn_stats<<<C, 256, 0, c.st>>>(src, E3, (long)C * E3, 2, E3, c.mean + cofs, c.rstd + cofs);
    const long tot = (long)2 * C * E3;
    k_bn_apply<<<(unsigned)((tot + 255) / 256), 256, 0, c.st>>>(
        src, dst, c.mean + cofs, c.rstd + cofs, g + cofs, bb + cofs, C, Cdst, cofs, tot);
}

} // namespace

extern "C" void kernel_launch(void* const* d_in, const int* in_sizes, int n_in,
                              void* d_out, int out_size, void* d_ws, size_t ws_size,
                              hipStream_t stream)
{
    const float* x      = (const float*)d_in[0];   // (2,3,64,64)
    const float* k_s2   = (const float*)d_in[1];   // (3,25,64)
    const float* bias_s2= (const float*)d_in[2];
    const float* k1     = (const float*)d_in[3];   // (25,25,64)
    const float* bias1  = (const float*)d_in[4];
    const float* k2     = (const float*)d_in[5];   // (50,25,64)
    const float* bias2  = (const float*)d_in[6];
    const float* k3     = (const float*)d_in[7];   // (75,36,64)
    const float* bias3  = (const float*)d_in[8];
    const float* g1 = (const float*)d_in[9],  *b1 = (const float*)d_in[10];
    const float* g2 = (const float*)d_in[11], *b2 = (const float*)d_in[12];
    const float* g3 = (const float*)d_in[13], *b3 = (const float*)d_in[14];
    const float* g4 = (const float*)d_in[15], *b4 = (const float*)d_in[16];
    float* out = (float*)d_out;
    (void)in_sizes; (void)n_in; (void)out_size; (void)ws_size;

    // ---- workspace layout (deterministic, recomputed each call) ----
    char* p0 = (char*)d_ws;
    size_t cur = 0;
    auto allocB = [&](size_t bytes) -> void* {
        void* r = p0 + cur;
        cur += (bytes + 255) & ~(size_t)255;
        return r;
    };
    double* lf  = (double*)allocB(65 * sizeof(double));
    float*  qw  = (float*)allocB(64 * 4);
    Ctx c; c.st = stream;
    c.WCre = (float*)allocB(4032 * 4);
    c.WCim = (float*)allocB(4032 * 4);
    c.VCre = (float*)allocB(4032 * 4);
    c.VCim = (float*)allocB(4032 * 4);
    c.DEQ  = (float*)allocB((size_t)32 * SZ2 * 4);
    c.mean = (float*)allocB(128 * 4);
    c.rstd = (float*)allocB(128 * 4);
    float* Kh   = (float*)allocB((size_t)2 * 75 * 36 * 63 * 4);   // worst-case Kh planes
    float* Xcs  = (float*)allocB((size_t)2 * 24192 * 4);          // s2 spectra (2,3,64,63)
    float* Fs   = (float*)allocB((size_t)2 * 12096 * 4);          // s2 F (2,3,32,63)
    c.WDF = (float*)allocB((size_t)64 * 32 * SZ2 * 4);
    c.WDI = (float*)allocB((size_t)64 * 32 * SZ2 * 4);
    float* x1   = (float*)allocB((size_t)2 * 25 * E3 * 4);
    float* x2   = (float*)allocB((size_t)2 * 25 * E3 * 4);
    float* x3   = (float*)allocB((size_t)2 * 25 * E3 * 4);
    float* Hcat = (float*)allocB((size_t)2 * 75 * E3 * 4);
    float* h4   = (float*)allocB((size_t)2 * 36 * E3 * 4);
    c.RA = (float*)allocB((size_t)77414400 * 4);   // T1 / C / Xt
    c.RB = (float*)allocB((size_t)76204800 * 4);   // Xc / z / Y
    c.RC = (float*)allocB((size_t)38102400 * 4);   // F / z_s2

    // ---- constant tables ----
    k_tables<<<1, 256, 0, stream>>>(lf, qw, c.WCre, c.WCim, c.VCre, c.VCim);
    k_wigner<<<dim3(16, 32, 65), 256, 0, stream>>>(lf, qw, c.WDF, c.WDI, c.DEQ);

    const float S2S = 1.0f / sqrtf(196608.f);      // 1/sqrt(NB*3*NL^2)

    // ================= s2_conv =================
    const long PXs = 24192, PFs = 12096, PKs = 4725;
    // Xc = fft_alpha(x), centered  (real x complex; rows = b*c*z = 384)
    gemm(stream, x, 64, 1, 0, 0, c.WCre, 63, 1, 0, 0, Xcs,       63, 1, 0, 0, 384, 63, 64, 1, 1, 1.f, 0);
    gemm(stream, x, 64, 1, 0, 0, c.WCim, 63, 1, 0, 0, Xcs + PXs, 63, 1, 0, 0, 384, 63, 64, 1, 1, 1.f, 0);
    // F[b,i,l,m] = sum_z WD0[z,l,m] * Xc[b,i,z,m]   (batch over m; WD0 = WDF[...,m_idx=31])
    for (int p = 0; p < 2; ++p)
        gemm(stream, Xcs + p * PXs, (long)64 * 63, 63, 1, 0,
                     c.WDF + 31, (long)32 * SZ2, SZ2, 63, 0,
                     Fs + p * PFs, (long)32 * 63, 63, 1, 0,
                     6, 32, 64, 63, 1, 1.f, 0);
    // Kh_s2 = S2_SCALE * k_s2 @ conj(PHASE)  (conj(PHASE) == WC)
    gemm(stream, k_s2, 64, 1, 0, 0, c.WCre, 63, 1, 0, 0, Kh,       63, 1, 0, 0, 75, 63, 64, 1, 1, S2S, 0);
    gemm(stream, k_s2, 64, 1, 0, 0, c.WCim, 63, 1, 0, 0, Kh + PKs, 63, 1, 0, 0, 75, 63, 64, 1, 1, S2S, 0);
    // z_s2[b,o,l,m,n] = sum_i F*Kh*D0EQ   -> RC
    {
        const long tot = 6350400;
        k_s2combine<<<(unsigned)((tot + 255) / 256), 256, 0, stream>>>(Fs, Kh, c.DEQ, c.RC);
    }
    so3_synth(c, c.RC, 25, x1);
    {
        const long tot = (long)2 * 25 * E3;
        k_biasadd<<<(unsigned)((tot + 255) / 256), 256, 0, stream>>>(x1, bias_s2, 25, tot);
    }

    // ================= block 1: relu(bn(x1)) -> conv1 -> x2 =================
    bn_block(c, x1, 25, 0, Hcat, 25, g1, b1);
    so3_conv(c, Hcat, 25, k1, 25, 1.0f / sqrtf(64.f * 25.f), Kh, x2, bias1);

    // ================= block 2: relu(bn(cat(x1,x2))) -> conv2 -> x3 =========
    bn_block(c, x1, 25, 0,  Hcat, 50, g2, b2);
    bn_block(c, x2, 25, 25, Hcat, 50, g2, b2);
    so3_conv(c, Hcat, 50, k2, 25, 1.0f / sqrtf(64.f * 50.f), Kh, x3, bias2);

    // ================= block 3: relu(bn(cat(x1,x2,x3))) -> conv3 -> h4 ======
    bn_block(c, x1, 25, 0,  Hcat, 75, g3, b3);
    bn_block(c, x2, 25, 25, Hcat, 75, g3, b3);
    bn_block(c, x3, 25, 50, Hcat, 75, g3, b3);
    so3_conv(c, Hcat, 75, k3, 36, 1.0f / sqrtf(64.f * 75.f), Kh, h4, bias3);

    // ================= final: relu(bn(h4)) then mean over alpha =============
    bn_block(c, h4, 36, 0, h4, 36, g4, b4);
    {
        const long tot = (long)2 * 36 * 64 * 64;
        k_meanlast<<<(unsigned)((tot + 255) / 256), 256, 0, stream>>>(h4, out, tot);
    }
}